// ChunkEmbedding_6133213299308
// MI455X (gfx1250) — compile-verified
//
#include <hip/hip_runtime.h>

// ---------------------------------------------------------------------------
// ChunkEmbedding for gfx1250 (MI455X).
// chunk[n,d] = sum_l (w[n,l]/sum_l w[n,l]) * E[ids[n,l], d]   via f32 WMMA
// then ragged scatter into ret[B, t, D] (+ CLS/SEP rows, mask).
// ---------------------------------------------------------------------------

typedef float v2f __attribute__((ext_vector_type(2)));
typedef float v8f __attribute__((ext_vector_type(8)));

#define CHUNK_LEN   64
#define DIM         768
#define CLS_IDX     101
#define SEP_IDX     102
#define WAVES       8
#define BLOCK       (WAVES * 32)
#define NTILES      (DIM / 16)          // 48
#define NT_PER_WAVE (NTILES / WAVES)    // 6

// ---------------- small helper kernels (segment bookkeeping) ----------------

__global__ void k_zero_counts(int* counts, const int* pB) {
    int b = threadIdx.x;
    if (b < *pB) counts[b] = 0;
}

__global__ void k_count(const int* __restrict__ map_ids, int n, int* counts) {
    int i = blockIdx.x * blockDim.x + threadIdx.x;
    if (i < n) atomicAdd(&counts[map_ids[i]], 1);
}

__global__ void k_offsets(const int* __restrict__ counts, int* offsets, const int* pB) {
    if (threadIdx.x == 0) {
        int B = *pB;
        int run = 0;
        for (int b = 0; b < B; ++b) { offsets[b] = run; run += counts[b]; }
    }
}

// ------------- init: zero ret, write CLS/SEP rows, write mask ---------------

__global__ void k_init(float* __restrict__ out, long total,
                       const float* __restrict__ emb,
                       const int* __restrict__ counts,
                       const int* pB, const int* pMM) {
    long i = (long)blockIdx.x * blockDim.x + threadIdx.x;
    if (i >= total) return;
    int B = *pB;
    int t = *pMM + 2;
    long retElems = (long)B * t * DIM;
    if (i < retElems) {
        int  d = (int)(i % DIM);
        long r = i / DIM;
        int  j = (int)(r % t);
        int  b = (int)(r / t);
        float v = 0.0f;
        if (j == 0)                 v = emb[(long)CLS_IDX * DIM + d];
        else if (j == counts[b] + 1) v = emb[(long)SEP_IDX * DIM + d];
        out[i] = v;
    } else {
        long k = i - retElems;
        int  j = (int)(k % t);
        int  b = (int)(k / t);
        out[i] = (j <= counts[b] + 1) ? 1.0f : 0.0f;
    }
}

// ---------------- main kernel: 16 chunks per workgroup, WMMA ----------------

__global__ __launch_bounds__(BLOCK)
void k_chunks(const int*   __restrict__ ids,
              const float* __restrict__ w,
              const int*   __restrict__ map_ids,
              const float* __restrict__ emb,
              const int*   __restrict__ offsets,
              const int*   pMM,
              float*       __restrict__ out,
              int n_chunks) {
    __shared__ float lds_w[16 * CHUNK_LEN];   // normalized weights, 16 chunks
    __shared__ int   lds_id[16 * CHUNK_LEN];  // token ids, 16 chunks
    __shared__ float lds_sum[16];

    const int tid  = threadIdx.x;
    const int lane = tid & 31;
    const int wave = tid >> 5;
    const int G    = blockIdx.x * 16;         // first chunk of this group

    // ---- stage 0: load weights+ids, compute per-chunk sum, normalize ----
    {
        int c = tid >> 4;        // chunk slot 0..15
        int j = tid & 15;        // handles tokens j*4 .. j*4+3
        int g = G + c;
        float wl[4]; int il[4];
        if (g < n_chunks) {
            const float* wp = w   + (long)g * CHUNK_LEN + j * 4;
            const int*   ip = ids + (long)g * CHUNK_LEN + j * 4;
#pragma unroll
            for (int q = 0; q < 4; ++q) { wl[q] = wp[q]; il[q] = ip[q]; }
        } else {
#pragma unroll
            for (int q = 0; q < 4; ++q) { wl[q] = 0.0f; il[q] = 0; }
        }
        if (j == 0) lds_sum[c] = 0.0f;
        __syncthreads();
        atomicAdd(&lds_sum[c], wl[0] + wl[1] + wl[2] + wl[3]);  // ds_add_f32
        __syncthreads();
        float denom = lds_sum[c];
#pragma unroll
        for (int q = 0; q < 4; ++q) {
            lds_w [c * CHUNK_LEN + j * 4 + q] = (denom != 0.0f) ? wl[q] / denom : 0.0f;
            lds_id[c * CHUNK_LEN + j * 4 + q] = il[q];
        }
        __syncthreads();
    }

    // ---- stage 1: accumulate D += A*B with one-hot-row A (row = chunk c) ----
    // A 16x4 f32 (2 VGPRs): lanes 0-15 hold K=0,1 ; lanes 16-31 hold K=2,3
    // B 4x16 f32 (2 VGPRs): VGPR0 -> K=koff, VGPR1 -> K=koff+1 (koff=2*laneHalf)
    v8f acc[NT_PER_WAVE] = {};

    const int laneHalf = lane >> 4;   // 0 or 1
    const int nl       = lane & 15;   // N index within tile / M row within half
    const int koff     = laneHalf * 2;

    for (int c = 0; c < 16; ++c) {
        const bool mine = (nl == c);
        for (int kb = 0; kb < CHUNK_LEN / 4; ++kb) {
            int base = c * CHUNK_LEN + kb * 4 + koff;
            v2f wp = *(const v2f*)&lds_w[base];    // ds_load_b64
            int2 ip = *(const int2*)&lds_id[base]; // ds_load_b64
            v2f a;
            a.x = mine ? wp.x : 0.0f;
            a.y = mine ? wp.y : 0.0f;
            const float* r0 = emb + (long)ip.x * DIM;
            const float* r1 = emb + (long)ip.y * DIM;
#pragma unroll
            for (int i = 0; i < NT_PER_WAVE; ++i) {
                int d0 = (wave * NT_PER_WAVE + i) * 16;
                v2f b;
                b.x = r0[d0 + nl];
                b.y = r1[d0 + nl];
                acc[i] = __builtin_amdgcn_wmma_f32_16x16x4_f32(
                    /*neg_a=*/false, a, /*neg_b=*/false, b,
                    /*c_mod=*/(short)0, acc[i],
                    /*reuse_a=*/false, /*reuse_b=*/false);
            }
        }
    }

    // ---- stage 2: scatter C tile rows to ret[map_id, pos+1, :] ----
    // C/D layout: VGPR v -> M = v + 8*laneHalf, N = nl
    const int t = *pMM + 2;
#pragma unroll
    for (int v = 0; v < 8; ++v) {
        int m = v + laneHalf * 8;
        int g = G + m;
        if (g < n_chunks) {
            int  mid = map_ids[g];
            long row = (long)mid * t + (g - offsets[mid]) + 1;
            float* orow = out + row * DIM;
#pragma unroll
            for (int i = 0; i < NT_PER_WAVE; ++i) {
                int d0 = (wave * NT_PER_WAVE + i) * 16;
                orow[d0 + nl] = acc[i][v];
            }
        }
    }
}

// ---------------------------------------------------------------------------

extern "C" void kernel_launch(void* const* d_in, const int* in_sizes, int n_in,
                              void* d_out, int out_size, void* d_ws, size_t ws_size,
                              hipStream_t stream) {
    const int*   input_ids = (const int*)  d_in[0];
    const float* kp_w      = (const float*)d_in[1];
    const int*   map_ids   = (const int*)  d_in[2];
    const float* emb       = (const float*)d_in[3];
    const int*   pB        = (const int*)  d_in[4];   // batch_size (device scalar)
    const int*   pMM       = (const int*)  d_in[5];   // max_map_len (device scalar)
    float*       out       = (float*)d_out;

    const int n = in_sizes[0] / CHUNK_LEN;            // number of chunks (4096)

    int* counts  = (int*)d_ws;
    int* offsets = (int*)((char*)d_ws + 4096);

    k_zero_counts<<<1, 1024, 0, stream>>>(counts, pB);
    k_count<<<(n + 255) / 256, 256, 0, stream>>>(map_ids, n, counts);
    k_offsets<<<1, 32, 0, stream>>>(counts, offsets, pB);

    long total = (long)out_size;
    k_init<<<(unsigned)((total + 255) / 256), 256, 0, stream>>>(
        out, total, emb, counts, pB, pMM);

    k_chunks<<<(n + 15) / 16, BLOCK, 0, stream>>>(
        input_ids, kp_w, map_ids, emb, offsets, pMM, out, n);
}